// IrregularPatchEmbed_49452253446282
// MI455X (gfx1250) — compile-verified
//
#include <hip/hip_runtime.h>
#include <hip/hip_bf16.h>
#include <math.h>

#define Bn 32
#define Tn 16384
#define Cn 128
#define Dn 256
#define Pn 2048
#define PATCHW 7.0f

typedef __attribute__((ext_vector_type(2))) float v2f;
typedef __attribute__((ext_vector_type(8))) float v8f;

// ---------------------------------------------------------------------------
// Phase 1: per-row cumsum -> patch id -> group rank -> group start offsets
// ---------------------------------------------------------------------------
__global__ __launch_bounds__(256) void ipe_scan_kernel(const float* __restrict__ td,
                                                       int* __restrict__ starts,
                                                       int* __restrict__ ngroups) {
  const int b   = blockIdx.x;
  const int tid = threadIdx.x;
  __shared__ float sf[256];
  __shared__ int   si[256];
  __shared__ float carry_t;
  __shared__ int   carry_pid;
  __shared__ int   carry_rank;
  if (tid == 0) { carry_t = 0.0f; carry_pid = -2147483647; carry_rank = -1; }
  __syncthreads();

  const float* row  = td + (size_t)b * Tn;
  int*         srow = starts + (size_t)b * (Tn + 1);

  for (int base = 0; base < Tn; base += 256) {
    float v = row[base + tid];
    // inclusive block scan (float cumsum)
    sf[tid] = v; __syncthreads();
    for (int off = 1; off < 256; off <<= 1) {
      float add = (tid >= off) ? sf[tid - off] : 0.0f;
      __syncthreads();
      sf[tid] += add;
      __syncthreads();
    }
    float t   = carry_t + sf[tid];
    int   pid = (int)floorf(t / PATCHW);

    si[tid] = pid; __syncthreads();
    int prev  = (tid == 0) ? carry_pid : si[tid - 1];
    int isnew = (pid != prev) ? 1 : 0;
    __syncthreads();

    // inclusive block scan (int) of new-group flags
    si[tid] = isnew; __syncthreads();
    for (int off = 1; off < 256; off <<= 1) {
      int add = (tid >= off) ? si[tid - off] : 0;
      __syncthreads();
      si[tid] += add;
      __syncthreads();
    }
    int rank = carry_rank + si[tid];
    if (isnew) srow[rank] = base + tid;
    __syncthreads();
    if (tid == 255) { carry_t = t; carry_pid = pid; carry_rank = rank; }
    __syncthreads();
  }
  if (tid == 0) {
    int ng = carry_rank + 1;
    ngroups[b] = ng;
    srow[ng]   = Tn;  // sentinel end of last group
  }
}

// ---------------------------------------------------------------------------
// Phase 2: segment means -> feats [B,P,C] in workspace, plus valid mask
// block = (p, b), 128 threads = one per channel
// ---------------------------------------------------------------------------
__global__ __launch_bounds__(128) void ipe_means_kernel(const float* __restrict__ x,
                                                        const int* __restrict__ starts,
                                                        const int* __restrict__ ngroups,
                                                        float* __restrict__ feats,
                                                        float* __restrict__ valid_out) {
  const int p = blockIdx.x;
  const int b = blockIdx.y;
  const int c = threadIdx.x;
  const int ng = ngroups[b];
  const int g  = ng - Pn + p;
  const size_t fo = ((size_t)b * Pn + p) * Cn + c;
  if (g < 0) {
    feats[fo] = 0.0f;
    if (c == 0) valid_out[(size_t)b * Pn + p] = 0.0f;
    return;
  }
  const int* srow = starts + (size_t)b * (Tn + 1);
  const int s = srow[g];
  const int e = srow[g + 1];
  const float* xr = x + ((size_t)b * Tn + s) * Cn + c;
  float acc = 0.0f;
  for (int t = s; t < e; ++t) { acc += *xr; xr += Cn; }
  feats[fo] = acc / (float)(e - s);
  if (c == 0) valid_out[(size_t)b * Pn + p] = 1.0f;
}

// ---------------------------------------------------------------------------
// Phase 3: feats[B,P,C] @ W[C,D] + bias via V_WMMA_F32_16X16X4_F32
// One wave per 64x16 output tile (4 p-subtiles share each B load pair),
// 8 waves per block. Inner-loop ratio: 4 WMMA : (4 b64 + 2 b32) loads.
// ---------------------------------------------------------------------------
__global__ __launch_bounds__(256) void ipe_gemm_kernel(const float* __restrict__ feats,
                                                       const float* __restrict__ Wm,
                                                       const float* __restrict__ bias,
                                                       float* __restrict__ out) {
  const int lane = threadIdx.x & 31;
  const int half = lane >> 4;   // 0: K pair (k,k+1)   1: K pair (k+2,k+3)
  const int lm   = lane & 15;   // M index (A) / N index (B)

  int tile = blockIdx.x * 8 + (threadIdx.x >> 5);
  const int dt  = tile & 15; tile >>= 4;   // 16 d-tiles of 16  (D=256)
  const int pt4 = tile & 31; tile >>= 5;   // 32 p-tiles of 64  (P=2048)
  const int bb  = tile;                    // batch
  const int d0  = dt * 16;
  const int p0  = pt4 * 64;

  // A: lane lm holds feats row (p0 + sub*16 + lm), K = k + 2*half + {0,1}
  const float* arow = feats + ((size_t)bb * Pn + (size_t)p0 + lm) * Cn + 2 * half;
  // B: lane lm holds W column (d0+lm), K = k + 2*half + {0,1}
  const float* bcol = Wm + (size_t)(2 * half) * Dn + d0 + lm;

  v8f acc0 = {}, acc1 = {}, acc2 = {}, acc3 = {};
  #pragma unroll 4
  for (int k = 0; k < Cn; k += 4) {
    v2f bv;
    bv.x = bcol[(size_t)k * Dn];
    bv.y = bcol[(size_t)k * Dn + Dn];
    v2f a0 = *(const v2f*)(arow + k);
    v2f a1 = *(const v2f*)(arow + (size_t)16 * Cn + k);
    v2f a2 = *(const v2f*)(arow + (size_t)32 * Cn + k);
    v2f a3 = *(const v2f*)(arow + (size_t)48 * Cn + k);
    acc0 = __builtin_amdgcn_wmma_f32_16x16x4_f32(false, a0, false, bv, (short)0, acc0, false, false);
    acc1 = __builtin_amdgcn_wmma_f32_16x16x4_f32(false, a1, false, bv, (short)0, acc1, false, false);
    acc2 = __builtin_amdgcn_wmma_f32_16x16x4_f32(false, a2, false, bv, (short)0, acc2, false, false);
    acc3 = __builtin_amdgcn_wmma_f32_16x16x4_f32(false, a3, false, bv, (short)0, acc3, false, false);
  }

  const float badd = bias[d0 + lm];
  // C/D layout: VGPR i, lanes 0-15 -> row i, lanes 16-31 -> row i+8; col = lane&15
  float* orow = out + ((size_t)bb * Pn + (size_t)p0 + half * 8) * Dn + d0 + lm;
  #pragma unroll
  for (int i = 0; i < 8; ++i) {
    orow[(size_t)i * Dn]                        = acc0[i] + badd;
    orow[(size_t)(16 + i) * Dn]                 = acc1[i] + badd;
    orow[(size_t)(32 + i) * Dn]                 = acc2[i] + badd;
    orow[(size_t)(48 + i) * Dn]                 = acc3[i] + badd;
  }
}

// ---------------------------------------------------------------------------
extern "C" void kernel_launch(void* const* d_in, const int* in_sizes, int n_in,
                              void* d_out, int out_size, void* d_ws, size_t ws_size,
                              hipStream_t stream) {
  const float* x    = (const float*)d_in[0];  // [B,T,C]
  const float* td   = (const float*)d_in[1];  // [B,T]
  const float* Wm   = (const float*)d_in[2];  // [C,D]
  const float* bias = (const float*)d_in[3];  // [D]

  float* patches = (float*)d_out;                       // [B,P,D]
  float* valid   = patches + (size_t)Bn * Pn * Dn;      // [B,P]

  // workspace layout
  int*   starts  = (int*)d_ws;                          // B*(T+1)
  int*   ngroups = starts + (size_t)Bn * (Tn + 1);      // B
  float* feats   = (float*)(ngroups + Bn);              // B*P*C (16B aligned)

  ipe_scan_kernel<<<Bn, 256, 0, stream>>>(td, starts, ngroups);

  dim3 g2(Pn, Bn);
  ipe_means_kernel<<<g2, Cn, 0, stream>>>(x, starts, ngroups, feats, valid);

  const int tiles = Bn * (Pn / 64) * (Dn / 16);         // 16384 waves (64x16 tiles)
  ipe_gemm_kernel<<<tiles / 8, 256, 0, stream>>>(feats, Wm, bias, patches);
}